// MultiLevelHashEncoding_11570641896045
// MI455X (gfx1250) — compile-verified
//
#include <hip/hip_runtime.h>
#include <math.h>

#define N_ENC     16384
#define HASH_MASK 16383u
#define N_LEVELS  16
#define P2        2654435761u
#define P3        805459861u
#define TPB       256
#define CHUNK     8192

typedef __attribute__((address_space(1))) int as1_int;
typedef __attribute__((address_space(3))) int as3_int;

struct ResParams { int res[N_LEVELS]; };

struct DimTerms { unsigned int h0, h1; float w0, w1; };

__device__ __forceinline__ DimTerms make_dim(float q, float rf, int res, unsigned int prime) {
    // align_corners=False unnormalization, identical op order to reference
    float pos = ((q + 1.0f) * rf - 1.0f) * 0.5f;
    float f0  = floorf(pos);
    float fr  = pos - f0;
    int   i0  = (int)f0;
    DimTerms d;
    d.h0 = (unsigned int)i0 * prime;
    d.h1 = (unsigned int)(i0 + 1) * prime;
    bool v0 = (i0 >= 0)     && (i0 < res);
    bool v1 = (i0 + 1 >= 0) && (i0 + 1 < res);
    // fold the "valid" (zero-padding) mask into the per-dim weight
    d.w0 = v0 ? (1.0f - fr) : 0.0f;
    d.w1 = v1 ? fr          : 0.0f;
    return d;
}

__global__ __launch_bounds__(TPB) void hashgrid_enc_kernel(
    const float* __restrict__ x,       // (B,3)
    const float* __restrict__ emb,     // (E=2, N_ENC, L=16)
    float* __restrict__ out,           // (B, L, 2)
    int B, ResParams rp)
{
    __shared__ float2 tab[N_ENC];      // 128 KB: this level's table as (feat0,feat1) pairs

    const int level = blockIdx.y;
    const int res   = rp.res[level];
    const int tid   = threadIdx.x;

    // ---- Stage level table into LDS, transposed: tab[n] = {emb[0][n][l], emb[1][n][l]} ----
    // element (e,n,l) lives at (e*N_ENC + n)*N_LEVELS + l ; feature planes are 1 MB apart.
    const float* __restrict__ gsrc = emb + level;
#if __has_builtin(__builtin_amdgcn_global_load_async_to_lds_b32)
    {
        const float* g0 = gsrc + (size_t)tid * N_LEVELS;                    // feat 0
        const float* g1 = gsrc + ((size_t)N_ENC + (size_t)tid) * N_LEVELS; // feat 1
        float*       l0 = (float*)&tab[tid];
#pragma unroll 4
        for (int it = 0; it < N_ENC / TPB; ++it) {   // fixed trip count: 64
            __builtin_amdgcn_global_load_async_to_lds_b32(
                (as1_int*)(void*)g0, (as3_int*)(void*)l0, 0, 0);
            __builtin_amdgcn_global_load_async_to_lds_b32(
                (as1_int*)(void*)g1, (as3_int*)(void*)(l0 + 1), 0, 0);
            g0 += (size_t)TPB * N_LEVELS;
            g1 += (size_t)TPB * N_LEVELS;
            l0 += 2 * TPB;
        }
        asm volatile("s_wait_asynccnt 0" ::: "memory");
    }
#else
    for (int it = 0; it < N_ENC / TPB; ++it) {
        int n = tid + it * TPB;
        float2 v;
        v.x = gsrc[(size_t)n * N_LEVELS];
        v.y = gsrc[((size_t)N_ENC + (size_t)n) * N_LEVELS];
        tab[n] = v;
    }
#endif
    __syncthreads();

    // ---- Process this block's chunk of points for this level ----
    const int   base = blockIdx.x * CHUNK;
    const int   stop = (base + CHUNK < B) ? (base + CHUNK) : B;
    const float rf   = (float)res;

#pragma unroll 2
    for (int b = base + tid; b < stop; b += TPB) {
        float qx = x[3 * b + 0];
        float qy = x[3 * b + 1];
        float qz = x[3 * b + 2];

        DimTerms dx = make_dim(qx, rf, res, P3);   // query dim0 -> prime 805459861
        DimTerms dy = make_dim(qy, rf, res, P2);   // query dim1 -> prime 2654435761
        DimTerms dz = make_dim(qz, rf, res, 1u);   // query dim2 -> prime 1

        float accx = 0.0f, accy = 0.0f;
#pragma unroll
        for (int c = 0; c < 8; ++c) {
            unsigned int hx = (c & 4) ? dx.h1 : dx.h0;
            unsigned int hy = (c & 2) ? dy.h1 : dy.h0;
            unsigned int hz = (c & 1) ? dz.h1 : dz.h0;
            float        w  = ((c & 4) ? dx.w1 : dx.w0)
                            * ((c & 2) ? dy.w1 : dy.w0)
                            * ((c & 1) ? dz.w1 : dz.w0);
            unsigned int h  = (hx ^ hy ^ hz) & HASH_MASK;  // == ref's masked chain
            float2 v = tab[h];                             // ds_load_b64 gather
            accx = fmaf(v.x, w, accx);
            accy = fmaf(v.y, w, accy);
        }

        float2 o; o.x = accx; o.y = accy;
        *(float2*)(out + (size_t)b * (N_LEVELS * 2) + level * 2) = o;
    }
}

extern "C" void kernel_launch(void* const* d_in, const int* in_sizes, int n_in,
                              void* d_out, int out_size, void* d_ws, size_t ws_size,
                              hipStream_t stream) {
    (void)n_in; (void)out_size; (void)d_ws; (void)ws_size;
    const float* x   = (const float*)d_in[0];   // (B,3) float32
    const float* emb = (const float*)d_in[1];   // (2, 16384, 16) float32
    float*       out = (float*)d_out;           // (B, 16, 2) float32
    const int B = in_sizes[0] / 3;

    // Replicate float32 ceil(exp(linspace(log(16), log(512), 16)))
    ResParams rp;
    {
        float lo = logf(16.0f), hi = logf(512.0f);
        float step = (hi - lo) / 15.0f;
        for (int i = 0; i < N_LEVELS; ++i) {
            float t = lo + step * (float)i;
            rp.res[i] = (int)ceilf(expf(t));
        }
    }

    dim3 grid((B + CHUNK - 1) / CHUNK, N_LEVELS, 1);
    dim3 block(TPB, 1, 1);
    hipLaunchKernelGGL(hashgrid_enc_kernel, grid, block, 0, stream, x, emb, out, B, rp);
}